// PrefetchDenseInstanceNorm_61804579389968
// MI455X (gfx1250) — compile-verified
//
#include <hip/hip_runtime.h>

// Problem constants (all powers of two -> index math is shifts/masks).
#define C_    128
#define H_    512
#define W_    512
#define HW_   (H_ * W_)        // 1 << 18
#define CHW_  (C_ * HW_)       // 1 << 25
#define TBL_H 12               // YA + 2
#define TBL_W 12               // XA + 2

static_assert((HW_ & (HW_ - 1)) == 0, "HW must be pow2");
static_assert(HW_ % 1024 == 0, "pass-2 block must not straddle channels");

typedef __attribute__((ext_vector_type(2))) float v2f;
typedef __attribute__((ext_vector_type(4))) float v4f;
typedef __attribute__((ext_vector_type(8))) float v8f;

// D = A(16x4,f32) x B(4x16,f32) + C(16x16,f32). With B == all-ones, every row of D
// accumulates the sum of that row of A (replicated across 16 columns). Sum is
// permutation-invariant, so arbitrary element placement in the A layout is fine.
__device__ __forceinline__ v8f wmma_sum_acc(v2f a, v2f ones, v8f c) {
  return __builtin_amdgcn_wmma_f32_16x16x4_f32(
      /*neg_a=*/false, a, /*neg_b=*/false, ones,
      /*c_mod=*/(short)0, c, /*reuse_a=*/false, /*reuse_b=*/false);
}

// ---------------------------------------------------------------------------
// Pass 1: one block per channel. WMMA-based Sigma(x), Sigma(x^2) over pre_x[c],
// then thread 0 finishes stats in fp64 and builds the 3x3 interp grids.
// ws layout per channel (20 floats): [mean, 1/std, mgrid[9], sinv_grid[9]]
// ---------------------------------------------------------------------------
__global__ __launch_bounds__(256) void pdin_stats_kernel(
    const float* __restrict__ x,       // (2,C,H,W); pre_x = x + CHW_
    const float* __restrict__ mtab,    // (C,12,12)
    const float* __restrict__ stab,    // (C,12,12)
    const int* __restrict__ ya_p, const int* __restrict__ xa_p,
    const int* __restrict__ pya_p, const int* __restrict__ pxa_p,
    float* __restrict__ ws)
{
  const int c = blockIdx.x;
  const float* px = x + (size_t)(C_ + c) * HW_;
  const int wave = threadIdx.x >> 5;
  const int lane = threadIdx.x & 31;
  const float* base = px + wave * 128 + lane * 4;   // 128 elems / wave / iter

  const v2f ones = {1.0f, 1.0f};
  v8f accS = {0, 0, 0, 0, 0, 0, 0, 0};
  v8f accQ = {0, 0, 0, 0, 0, 0, 0, 0};

  // 8 waves * 128 elems = 1024 elems per block per iter; HW_/1024 = 256 iters.
  for (int t = 0; t < HW_ / 1024; ++t) {
    v4f a = *(const v4f*)(base + t * 1024);         // keep temporal: reused in pass 2 (fits L2)
    v2f a0 = a.xy, a1 = a.zw;
    v2f q0 = a0 * a0, q1 = a1 * a1;
    accS = wmma_sum_acc(a0, ones, accS);
    accQ = wmma_sum_acc(q0, ones, accQ);
    accS = wmma_sum_acc(a1, ones, accS);
    accQ = wmma_sum_acc(q1, ones, accQ);
  }

  // Sum of all 256 D elements = 16 * (sum of accumulated inputs).
  float s = 0.0f, q = 0.0f;
#pragma unroll
  for (int i = 0; i < 8; ++i) { s += accS[i]; q += accQ[i]; }
#pragma unroll
  for (int off = 16; off > 0; off >>= 1) {
    s += __shfl_xor(s, off);
    q += __shfl_xor(q, off);
  }

  __shared__ float sh_s[8], sh_q[8];
  if (lane == 0) { sh_s[wave] = s * 0.0625f; sh_q[wave] = q * 0.0625f; }
  __syncthreads();

  if (threadIdx.x == 0) {
    double S = 0.0, Q = 0.0;
    for (int w = 0; w < 8; ++w) { S += (double)sh_s[w]; Q += (double)sh_q[w]; }
    const double N = (double)HW_;
    const double mean_d = S / N;
    const double var_d  = (Q - S * S / N) / (N - 1.0);   // ddof = 1
    const float meanf = (float)mean_d;
    const float stdv  = (float)sqrt(var_d);

    float* o = ws + c * 20;
    o[0] = meanf;
    o[1] = 1.0f / stdv;

    const int ya = ya_p[0], xa = xa_p[0];
    const int py = pya_p[0] + 1, pxx = pxa_p[0] + 1;  // updated table entry

    float m[9], sd[9];
#pragma unroll
    for (int j = 0; j < 3; ++j)
#pragma unroll
      for (int i = 0; i < 3; ++i) {
        const int ty = ya + j, tx = xa + i;
        float mv = mtab[(c * TBL_H + ty) * TBL_W + tx];
        float sv = stab[(c * TBL_H + ty) * TBL_W + tx];
        if (ty == py && tx == pxx) { mv = meanf; sv = stdv; }  // .at[].set() lands in window
        m[j * 3 + i] = mv;
        sd[j * 3 + i] = sv;
      }
    const float mc = m[4], sc = sd[4];   // window center AFTER substitution
#pragma unroll
    for (int k = 0; k < 9; ++k) {
      const float mv = (m[k]  == 0.0f) ? mc : m[k];
      const float sv = (sd[k] == 0.0f) ? sc : sd[k];
      o[2 + k]  = mv;
      o[11 + k] = 1.0f / sv;             // reference interpolates 1/s
    }
  }
}

// ---------------------------------------------------------------------------
// Pass 2: fused elementwise over (2,C,H,W), float4 per thread.
// img==1: pre-normalize with (mean, 1/std). img==0: 3x3 bilinear maps.
// Blocks are uniform in (img, c) -> scalar branch + scalar grid loads.
// ---------------------------------------------------------------------------
__global__ __launch_bounds__(256) void pdin_apply_kernel(
    const float* __restrict__ x,
    const float* __restrict__ wgt,     // (1,C,1,1)
    const float* __restrict__ bia,     // (1,C,1,1)
    const float* __restrict__ ws,
    float* __restrict__ out)
{
  const unsigned tid = blockIdx.x * 256u + threadIdx.x;
  const unsigned e   = tid * 4u;                                  // element index
  const int img = __builtin_amdgcn_readfirstlane((int)(e >> 25)); // block-uniform
  const unsigned rem = e & (CHW_ - 1u);
  const int c = __builtin_amdgcn_readfirstlane((int)(rem >> 18)); // block-uniform
  const unsigned p = rem & (HW_ - 1u);

  const float wc = wgt[c];
  const float bc = bia[c];
  const float* wsc = ws + c * 20;
  v4f r;

  if (img) {
    // pre_out = (pre_x - mean) / std * w + b
    const float mean = wsc[0];
    const float rstd = wsc[1];
    v4f v = *(const v4f*)(x + e);                  // temporal: served from L2 after pass 1
    r = (v - mean) * rstd * wc + bc;
  } else {
    v4f v = __builtin_nontemporal_load((const v4f*)(x + e));  // read-once stream
    const int y  = (int)(p >> 9);
    const int x0 = (int)(p & 511u);
    // pos = (i+0.5)/512 + 0.5 in (0.5, 1.5): exact in fp32, floor == (pos>=1).
    const float posy = ((float)y + 0.5f) * (1.0f / 512.0f) + 0.5f;
    const int   iy = (posy >= 1.0f) ? 1 : 0;
    const float wy = posy - (float)iy;

    const float* mg = wsc + 2;
    const float* sg = wsc + 11;
    float rm[3], rs[3];
#pragma unroll
    for (int k = 0; k < 3; ++k) {                  // lerp along Y first (matches ref order)
      const float m0 = mg[iy * 3 + k], m1 = mg[iy * 3 + 3 + k];
      rm[k] = m0 + wy * (m1 - m0);
      const float s0 = sg[iy * 3 + k], s1 = sg[iy * 3 + 3 + k];
      rs[k] = s0 + wy * (s1 - s0);
    }
#pragma unroll
    for (int j = 0; j < 4; ++j) {
      const float posx = ((float)(x0 + j) + 0.5f) * (1.0f / 512.0f) + 0.5f;
      const int   ix = (posx >= 1.0f) ? 1 : 0;
      const float wx = posx - (float)ix;
      const float mv = rm[ix] + wx * (rm[ix + 1] - rm[ix]);
      const float sv = rs[ix] + wx * (rs[ix + 1] - rs[ix]);
      r[j] = (v[j] - mv) * sv * wc + bc;
    }
  }
  __builtin_nontemporal_store(r, (v4f*)(out + e)); // write-once stream
}

extern "C" void kernel_launch(void* const* d_in, const int* in_sizes, int n_in,
                              void* d_out, int out_size, void* d_ws, size_t ws_size,
                              hipStream_t stream) {
  (void)in_sizes; (void)n_in; (void)out_size; (void)ws_size;
  const float* x    = (const float*)d_in[0];
  const float* mtab = (const float*)d_in[1];
  const float* stab = (const float*)d_in[2];
  const float* wgt  = (const float*)d_in[3];
  const float* bia  = (const float*)d_in[4];
  const int*   ya   = (const int*)d_in[5];
  const int*   xa   = (const int*)d_in[6];
  const int*   pya  = (const int*)d_in[7];
  const int*   pxa  = (const int*)d_in[8];
  float* out = (float*)d_out;
  float* ws  = (float*)d_ws;   // needs 128*20*4 = 10240 bytes

  pdin_stats_kernel<<<C_, 256, 0, stream>>>(x, mtab, stab, ya, xa, pya, pxa, ws);
  pdin_apply_kernel<<<(2 * CHW_) / 1024, 256, 0, stream>>>(x, wgt, bia, ws, out);
}